// AnchorFreeHead_24584392802325
// MI455X (gfx1250) — compile-verified
//
#include <hip/hip_runtime.h>
#include <hip/hip_bf16.h>
#include <math.h>

// Problem constants (match reference)
#define BB 8
#define TT 20000
#define DD 512
#define PP 128
#define CC 200
#define NNEG 3
#define HHID 256
#define NSEG (BB * PP)   // 1024

typedef float v2f __attribute__((ext_vector_type(2)));
typedef float v8f __attribute__((ext_vector_type(8)));

// ---------------------------------------------------------------------------
// Kernel 1: variable-length segment mean pooling (bandwidth-bound, ~270 MB).
// One block per segment; 128 threads, each owns 4 contiguous channels ->
// global_load_b128 per row per thread (512 B per wave per load).
// 2-row unroll with independent accumulators for ILP / latency hiding.
// ---------------------------------------------------------------------------
__global__ void seg_mean_kernel(const float* __restrict__ img,
                                const int* __restrict__ seg_start,
                                const int* __restrict__ seg_len,
                                float* __restrict__ vis) {
    const int seg = blockIdx.x;          // 0..NSEG-1  (b*P + p)
    const int b   = seg / PP;
    const int s   = seg_start[seg];
    const int L   = seg_len[seg];
    const int c4  = threadIdx.x;         // 0..127 -> channels [4*c4, 4*c4+3]

    const float4* base =
        (const float4*)(img + ((size_t)b * TT + s) * DD) + c4;
    const size_t rowstride4 = DD / 4;    // 128 float4 per row

    float4 a0 = make_float4(0.f, 0.f, 0.f, 0.f);
    float4 a1 = make_float4(0.f, 0.f, 0.f, 0.f);

    int t = 0;
    for (; t + 1 < L; t += 2) {
        const float4 r0 = base[(size_t)t * rowstride4];
        const float4 r1 = base[(size_t)(t + 1) * rowstride4];
        a0.x += r0.x; a0.y += r0.y; a0.z += r0.z; a0.w += r0.w;
        a1.x += r1.x; a1.y += r1.y; a1.z += r1.z; a1.w += r1.w;
    }
    if (t < L) {
        const float4 r0 = base[(size_t)t * rowstride4];
        a0.x += r0.x; a0.y += r0.y; a0.z += r0.z; a0.w += r0.w;
    }

    const float inv = 1.0f / (float)L;
    float4 o;
    o.x = (a0.x + a1.x) * inv;
    o.y = (a0.y + a1.y) * inv;
    o.z = (a0.z + a1.z) * inv;
    o.w = (a0.w + a1.w) * inv;
    ((float4*)(vis + (size_t)seg * DD))[c4] = o;
}

// ---------------------------------------------------------------------------
// Kernel 2: h = relu(vis @ W1 + b1)   [1024x512]@[512x256] via f32 WMMA.
// One wave computes one 16x16 tile of h. K-loop steps of 4.
// A fragment (16x4 f32): lane(0..15)=row M, vgpr j holds K = 2*(lane>>4)+j
// B fragment (4x16 f32): lane&15 = col N, vgpr j holds K = 2*(lane>>4)+j
// C/D (16x16 f32): vgpr r -> M = r + 8*(lane>>4), N = lane&15
// ---------------------------------------------------------------------------
__global__ void mlp_gemm1_wmma(const float* __restrict__ vis,
                               const float* __restrict__ W1,
                               const float* __restrict__ b1,
                               float* __restrict__ h) {
    const int gid  = blockIdx.x * blockDim.x + threadIdx.x;
    const int wave = gid >> 5;
    const int lane = threadIdx.x & 31;
    const int NT   = HHID / 16;            // 16 tiles along N
    const int tm   = wave / NT;
    const int tn   = wave % NT;
    const int m0   = tm * 16;
    const int n0   = tn * 16;
    const int hi   = lane >> 4;
    const int ln   = lane & 15;

    const float* arow = vis + (size_t)(m0 + ln) * DD;  // A row for this lane
    const float* bcol = W1 + n0 + ln;                  // B column for this lane

    v8f c = {};
    for (int k = 0; k < DD; k += 4) {
        const int ka = k + 2 * hi;
        v2f a;
        a.x = arow[ka];
        a.y = arow[ka + 1];
        v2f bmat;
        bmat.x = bcol[(size_t)ka * HHID];
        bmat.y = bcol[(size_t)(ka + 1) * HHID];
        c = __builtin_amdgcn_wmma_f32_16x16x4_f32(
            /*neg_a=*/false, a, /*neg_b=*/false, bmat,
            /*c_mod=*/(short)0, c, /*reuse_a=*/false, /*reuse_b=*/false);
    }

    const float bias = b1[n0 + ln];
    #pragma unroll
    for (int r = 0; r < 8; ++r) {
        float v = c[r] + bias;
        h[(size_t)(m0 + r + 8 * hi) * HHID + n0 + ln] = v > 0.0f ? v : 0.0f;
    }
}

// ---------------------------------------------------------------------------
// Kernel 3: out = vis + h @ W2 + b2   [1024x256]@[256x512] via f32 WMMA
// ---------------------------------------------------------------------------
__global__ void mlp_gemm2_wmma(const float* __restrict__ h,
                               const float* __restrict__ W2,
                               const float* __restrict__ b2,
                               const float* __restrict__ vis,
                               float* __restrict__ outp) {
    const int gid  = blockIdx.x * blockDim.x + threadIdx.x;
    const int wave = gid >> 5;
    const int lane = threadIdx.x & 31;
    const int NT   = DD / 16;              // 32 tiles along N
    const int tm   = wave / NT;
    const int tn   = wave % NT;
    const int m0   = tm * 16;
    const int n0   = tn * 16;
    const int hi   = lane >> 4;
    const int ln   = lane & 15;

    const float* arow = h + (size_t)(m0 + ln) * HHID;
    const float* bcol = W2 + n0 + ln;

    v8f c = {};
    for (int k = 0; k < HHID; k += 4) {
        const int ka = k + 2 * hi;
        v2f a;
        a.x = arow[ka];
        a.y = arow[ka + 1];
        v2f bmat;
        bmat.x = bcol[(size_t)ka * DD];
        bmat.y = bcol[(size_t)(ka + 1) * DD];
        c = __builtin_amdgcn_wmma_f32_16x16x4_f32(
            false, a, false, bmat, (short)0, c, false, false);
    }

    const float bias = b2[n0 + ln];
    #pragma unroll
    for (int r = 0; r < 8; ++r) {
        const size_t row = (size_t)(m0 + r + 8 * hi);
        const size_t off = row * DD + n0 + ln;
        outp[off] = vis[off] + c[r] + bias;
    }
}

// ---------------------------------------------------------------------------
// Kernel 4: per-segment contrastive loss.
// Normalization folded into logits: logits_k = (out . txt_k) / max(||out||,eps)
// One block (128 thr = 4 wave32) per segment; each thread owns one float4
// chunk (b128 loads, no loop); shuffle + LDS reduction.
// ---------------------------------------------------------------------------
__device__ __forceinline__ float dot4(const float4 a, const float4 b) {
    return a.x * b.x + a.y * b.y + a.z * b.z + a.w * b.w;
}

__global__ void loss_kernel(const float* __restrict__ outp,
                            const float* __restrict__ text,
                            const int* __restrict__ labels,
                            const int* __restrict__ neg_idx,
                            float* __restrict__ segloss) {
    const int seg = blockIdx.x;
    const int b   = seg / PP;
    const int tid = threadIdx.x;          // 0..127

    const int i0 = labels[seg];
    const int i1 = neg_idx[seg * NNEG + 0];
    const int i2 = neg_idx[seg * NNEG + 1];
    const int i3 = neg_idx[seg * NNEG + 2];

    const float4* o  = (const float4*)(outp + (size_t)seg * DD);
    const float4* t0 = (const float4*)(text + ((size_t)b * CC + i0) * DD);
    const float4* t1 = (const float4*)(text + ((size_t)b * CC + i1) * DD);
    const float4* t2 = (const float4*)(text + ((size_t)b * CC + i2) * DD);
    const float4* t3 = (const float4*)(text + ((size_t)b * CC + i3) * DD);

    const float4 ov = o[tid];
    float nn = dot4(ov, ov);
    float d0 = dot4(ov, t0[tid]);
    float d1 = dot4(ov, t1[tid]);
    float d2 = dot4(ov, t2[tid]);
    float d3 = dot4(ov, t3[tid]);

    // wave32 reduction
    for (int off = 16; off > 0; off >>= 1) {
        nn += __shfl_down(nn, off, 32);
        d0 += __shfl_down(d0, off, 32);
        d1 += __shfl_down(d1, off, 32);
        d2 += __shfl_down(d2, off, 32);
        d3 += __shfl_down(d3, off, 32);
    }

    __shared__ float sred[4][5];
    const int wv = tid >> 5;
    const int ll = tid & 31;
    if (ll == 0) {
        sred[wv][0] = nn; sred[wv][1] = d0; sred[wv][2] = d1;
        sred[wv][3] = d2; sred[wv][4] = d3;
    }
    __syncthreads();

    if (tid == 0) {
        float acc[5] = {0.f, 0.f, 0.f, 0.f, 0.f};
        for (int w = 0; w < 4; ++w)
            for (int q = 0; q < 5; ++q)
                acc[q] += sred[w][q];
        const float nrm = fmaxf(sqrtf(acc[0]), 1e-12f);
        const float inv = 1.0f / nrm;
        const float l0 = acc[1] * inv, l1 = acc[2] * inv;
        const float l2 = acc[3] * inv, l3 = acc[4] * inv;
        const float m  = fmaxf(fmaxf(l0, l1), fmaxf(l2, l3));
        const float lse = m + logf(expf(l0 - m) + expf(l1 - m) +
                                   expf(l2 - m) + expf(l3 - m));
        segloss[seg] = lse - l0;
    }
}

// ---------------------------------------------------------------------------
// Kernel 5: deterministic final mean over 1024 per-segment losses.
// ---------------------------------------------------------------------------
__global__ void reduce_loss_kernel(const float* __restrict__ segloss,
                                   float* __restrict__ loss) {
    __shared__ float s[256];
    const int tid = threadIdx.x;
    float a = 0.0f;
    for (int i = tid; i < NSEG; i += 256) a += segloss[i];
    s[tid] = a;
    __syncthreads();
    for (int st = 128; st > 0; st >>= 1) {
        if (tid < st) s[tid] += s[tid + st];
        __syncthreads();
    }
    if (tid == 0) loss[0] = s[0] * (1.0f / (float)NSEG);
}

// ---------------------------------------------------------------------------
extern "C" void kernel_launch(void* const* d_in, const int* in_sizes, int n_in,
                              void* d_out, int out_size, void* d_ws, size_t ws_size,
                              hipStream_t stream) {
    const float* img    = (const float*)d_in[0];
    const float* text   = (const float*)d_in[1];
    const float* W1     = (const float*)d_in[2];
    const float* b1     = (const float*)d_in[3];
    const float* W2     = (const float*)d_in[4];
    const float* b2     = (const float*)d_in[5];
    const int*   sstart = (const int*)d_in[6];
    const int*   slen   = (const int*)d_in[7];
    const int*   labels = (const int*)d_in[8];
    const int*   negidx = (const int*)d_in[9];
    float*       loss   = (float*)d_out;

    // Workspace layout (floats): vis[1024*512] | h[1024*256] | out[1024*512] | segloss[1024]
    float* vis     = (float*)d_ws;
    float* h       = vis  + (size_t)NSEG * DD;
    float* outp    = h    + (size_t)NSEG * HHID;
    float* segloss = outp + (size_t)NSEG * DD;

    // 1) segment mean pooling (dominant: ~270 MB streamed from HBM)
    hipLaunchKernelGGL(seg_mean_kernel, dim3(NSEG), dim3(128), 0, stream,
                       img, sstart, slen, vis);

    // 2) GEMM1: 1024x256 output -> 64*16 = 1024 tiles, 8 waves/block -> 128 blocks
    hipLaunchKernelGGL(mlp_gemm1_wmma, dim3((NSEG / 16) * (HHID / 16) / 8), dim3(256),
                       0, stream, vis, W1, b1, h);

    // 3) GEMM2: 1024x512 output -> 64*32 = 2048 tiles, 8 waves/block -> 256 blocks
    hipLaunchKernelGGL(mlp_gemm2_wmma, dim3((NSEG / 16) * (DD / 16) / 8), dim3(256),
                       0, stream, h, W2, b2, vis, outp);

    // 4) per-segment loss (normalization folded in)
    hipLaunchKernelGGL(loss_kernel, dim3(NSEG), dim3(128), 0, stream,
                       outp, text, labels, negidx, segloss);

    // 5) deterministic mean
    hipLaunchKernelGGL(reduce_loss_kernel, dim3(1), dim3(256), 0, stream,
                       segloss, loss);
}